// Neighborlist_62388694942378
// MI455X (gfx1250) — compile-verified
//
#include <hip/hip_runtime.h>
#include <cstdint>

// Neighborlist screening: gather -> diff -> norm -> (dummy & cutoff) mask.
// Memory-bound (AI ~0.2 FLOP/B) -> no matrix math, WMMA not applicable.
// MI455X strategy:
//  - coords (1.2MB) + species (0.4MB) stay resident in 192MB L2 -> gathers hit L2
//  - shift_values stream staged with CDNA5 async global->LDS DMA (ASYNCcnt),
//    overlapping the independent stream with the idx->gather dependency chains
//  - full tiles only in the main kernel (tail handled separately) so the LDS
//    path is the *only* shift producer and cannot be folded back to global

#define NL_CUTOFF 5.2f

typedef int v4i __attribute__((vector_size(16)));
typedef __attribute__((address_space(3))) float lds_f;
typedef __attribute__((address_space(3))) char  lds_c;
typedef __attribute__((address_space(1))) v4i   glb_v4i;
typedef __attribute__((address_space(3))) v4i   lds_v4i;

struct __align__(4) F3 { float x, y, z; };

// ---------------------------------------------------------------------------
// Main kernel: every block processes one FULL tile of 1024 pairs.
// ---------------------------------------------------------------------------
__global__ __launch_bounds__(256) void nl_screen_full(
    const int*   __restrict__ species,   // [N]
    const float* __restrict__ coords,    // [N*3]
    const int*   __restrict__ nbr,       // [2*P]
    const float* __restrict__ shift,     // [P*3]
    int*   __restrict__ out_idx,         // [2*P]
    float* __restrict__ out_dist,        // [P]
    float* __restrict__ out_diff,        // [P*3]
    int P, int opaque_zero)              // opaque_zero == 0 at runtime
{
    constexpr int BLK   = 256;
    constexpr int ITEMS = 4;
    constexpr int TILE  = BLK * ITEMS;          // 1024 pairs / block
    __shared__ float s_shift[TILE * 3];         // 12 KB

    const int tid      = threadIdx.x;
    const int tileBase = (int)blockIdx.x * TILE;

    // Opaque, never-executed store: ensures the LDS object has a def in IR.
    if (opaque_zero) s_shift[opaque_zero & 1023] = 1.0f;

    // --- Stage this tile's shift vectors (12288 B) into LDS via async DMA:
    //     each thread moves 48 contiguous bytes with 3x b128.
    {
        const int goff = tileBase * 12 + tid * 48;
#if __has_builtin(__builtin_amdgcn_global_load_async_to_lds_b128)
        glb_v4i* gsrc = (glb_v4i*)((const char*)shift + goff);
        lds_v4i* ldst = (lds_v4i*)((lds_c*)(lds_f*)&s_shift[0] + tid * 48);
        __builtin_amdgcn_global_load_async_to_lds_b128(gsrc + 0, ldst + 0, 0, 0);
        __builtin_amdgcn_global_load_async_to_lds_b128(gsrc + 1, ldst + 1, 0, 0);
        __builtin_amdgcn_global_load_async_to_lds_b128(gsrc + 2, ldst + 2, 0, 0);
#else
        unsigned ldsoff = (unsigned)(uintptr_t)(lds_f*)&s_shift[0] +
                          (unsigned)tid * 48u;
        asm volatile(
            "global_load_async_to_lds_b128 %0, %1, %2 offset:0\n\t"
            "global_load_async_to_lds_b128 %0, %1, %2 offset:16\n\t"
            "global_load_async_to_lds_b128 %0, %1, %2 offset:32"
            :
            : "v"(ldsoff), "v"((unsigned)goff), "s"(shift)
            : "memory");
#endif
#if __has_builtin(__builtin_amdgcn_s_wait_asynccnt)
        __builtin_amdgcn_s_wait_asynccnt(0);
#else
        asm volatile("s_wait_asynccnt 0x0" ::: "memory");
#endif
        __syncthreads();
    }

    // --- Launder the LDS base address so the consumer loads have opaque
    //     provenance: they cannot be value-forwarded from the async source
    //     nor folded to undef, and must lower to real ds_load instructions.
    unsigned lbase = (unsigned)(uintptr_t)(lds_f*)&s_shift[0];
    asm volatile("" : "+v"(lbase));
    const lds_f* lsh = (const lds_f*)(uintptr_t)lbase;

    const int* __restrict__ nbr0 = nbr;
    const int* __restrict__ nbr1 = nbr + P;

    #pragma unroll
    for (int k = 0; k < ITEMS; ++k) {
        const int q = k * BLK + tid;            // pair index within tile
        const int p = tileBase + q;             // global pair index

        // Coalesced streaming index loads
        const int i0 = nbr0[p];
        const int i1 = nbr1[p];

        // L2-resident gathers
        const bool dum0 = (species[i0] == -1);
        const bool dum1 = (species[i1] == -1);
        const F3 c0 = ((const F3*)coords)[i0];
        const F3 c1 = ((const F3*)coords)[i1];

        // LDS shift read; bank = (3*q)%64, stride 3 coprime with 64 -> no conflicts
        const float sx = lsh[3 * q + 0];
        const float sy = lsh[3 * q + 1];
        const float sz = lsh[3 * q + 2];

        const float dx = c0.x - c1.x + sx;
        const float dy = c0.y - c1.y + sy;
        const float dz = c0.z - c1.z + sz;
        const float d2 = dx * dx + dy * dy + dz * dz;
        const float dist = (d2 > 0.0f) ? sqrtf(d2) : 0.0f;

        const bool keep = !(dum0 | dum1) && (dist <= NL_CUTOFF);

        // Coalesced stores, flat in reference return order
        out_idx[p]     = keep ? i0 : -1;
        out_idx[P + p] = keep ? i1 : -1;
        out_dist[p]    = keep ? dist : 0.0f;
        F3 o;
        o.x = keep ? dx : 0.0f;
        o.y = keep ? dy : 0.0f;
        o.z = keep ? dz : 0.0f;
        ((F3*)out_diff)[p] = o;
    }
}

// ---------------------------------------------------------------------------
// Tail kernel: remaining P % 1024 pairs (never launched for harness sizes).
// ---------------------------------------------------------------------------
__global__ __launch_bounds__(256) void nl_screen_tail(
    const int*   __restrict__ species,
    const float* __restrict__ coords,
    const int*   __restrict__ nbr,
    const float* __restrict__ shift,
    int*   __restrict__ out_idx,
    float* __restrict__ out_dist,
    float* __restrict__ out_diff,
    int P, int start)
{
    const int p = start + (int)blockIdx.x * 256 + (int)threadIdx.x;
    if (p >= P) return;

    const int i0 = nbr[p];
    const int i1 = nbr[P + p];
    const bool dum0 = (species[i0] == -1);
    const bool dum1 = (species[i1] == -1);
    const F3 c0 = ((const F3*)coords)[i0];
    const F3 c1 = ((const F3*)coords)[i1];
    const F3 s  = ((const F3*)shift)[p];

    const float dx = c0.x - c1.x + s.x;
    const float dy = c0.y - c1.y + s.y;
    const float dz = c0.z - c1.z + s.z;
    const float d2 = dx * dx + dy * dy + dz * dz;
    const float dist = (d2 > 0.0f) ? sqrtf(d2) : 0.0f;

    const bool keep = !(dum0 | dum1) && (dist <= NL_CUTOFF);

    out_idx[p]     = keep ? i0 : -1;
    out_idx[P + p] = keep ? i1 : -1;
    out_dist[p]    = keep ? dist : 0.0f;
    F3 o;
    o.x = keep ? dx : 0.0f;
    o.y = keep ? dy : 0.0f;
    o.z = keep ? dz : 0.0f;
    ((F3*)out_diff)[p] = o;
}

extern "C" void kernel_launch(void* const* d_in, const int* in_sizes, int n_in,
                              void* d_out, int out_size, void* d_ws, size_t ws_size,
                              hipStream_t stream) {
    // setup_inputs order: species [1,N] i32, coordinates [1,N,3] f32,
    //                     neighbor_idxs [2,P] i32, shift_values [P,3] f32
    const int*   species = (const int*)  d_in[0];
    const float* coords  = (const float*)d_in[1];
    const int*   nbr     = (const int*)  d_in[2];
    const float* shift   = (const float*)d_in[3];

    const int P = in_sizes[2] / 2;

    // Outputs concatenated flat in return order:
    //   indices [2,P] int32 (raw bits) | distances [P] f32 | diff_vectors [P,3] f32
    int*   out_idx  = (int*)d_out;
    float* out_dist = (float*)d_out + (size_t)2 * P;
    float* out_diff = (float*)d_out + (size_t)3 * P;

    const int TILE  = 1024;
    const int nFull = P / TILE;          // 6250 full tiles for P = 6.4M
    const int rem   = P - nFull * TILE;  // 0 for harness sizes

    if (nFull > 0) {
        nl_screen_full<<<nFull, 256, 0, stream>>>(species, coords, nbr, shift,
                                                  out_idx, out_dist, out_diff,
                                                  P, /*opaque_zero=*/0);
    }
    if (rem > 0) {
        const int grid = (rem + 255) / 256;
        nl_screen_tail<<<grid, 256, 0, stream>>>(species, coords, nbr, shift,
                                                 out_idx, out_dist, out_diff,
                                                 P, nFull * TILE);
    }
}